// QChannelAttention_50405736186578
// MI455X (gfx1250) — compile-verified
//
#include <hip/hip_runtime.h>

typedef float v2f __attribute__((ext_vector_type(2)));
typedef float v4f __attribute__((ext_vector_type(4)));
typedef float v8f __attribute__((ext_vector_type(8)));

#define B_ 16
#define C_ 256
#define SP 4096              // 64*64 spatial positions
#define NP (B_ * C_)         // 4096 (b,ch) pairs

__device__ __forceinline__ v4f vmax4(v4f a, v4f b) {
  v4f r;
  r.x = fmaxf(a.x, b.x);
  r.y = fmaxf(a.y, b.y);
  r.z = fmaxf(a.z, b.z);
  r.w = fmaxf(a.w, b.w);
  return r;
}

// ---------------------------------------------------------------------------
// Kernel 1: per-(b,ch) max over 64*64 spatial positions, componentwise on the
// quaternion float4. One block per (b,ch); 256 threads x 16 float4 loads.
// Default (RT) temporal policy on loads leaves the tail of x hot in the
// 192MB L2 for the scale pass.
// ---------------------------------------------------------------------------
__global__ __launch_bounds__(256) void qca_maxpool(const float* __restrict__ x,
                                                   float* __restrict__ pooled) {
  const int p = blockIdx.x;                       // 0..NP-1
  const v4f* xp = (const v4f*)x + (size_t)p * SP;
  v4f m = xp[threadIdx.x];
#pragma unroll
  for (int i = 1; i < 16; ++i) m = vmax4(m, xp[threadIdx.x + i * 256]);

  __shared__ v4f red[256];
  red[threadIdx.x] = m;
  __syncthreads();
  for (int s = 128; s > 0; s >>= 1) {
    if (threadIdx.x < s) red[threadIdx.x] = vmax4(red[threadIdx.x], red[threadIdx.x + s]);
    __syncthreads();
  }
  if (threadIdx.x == 0) ((v4f*)pooled)[p] = red[0];
}

// ---------------------------------------------------------------------------
// Kernel 2 (single wave, 32 threads): squeeze + excite.
// Stage pooled (64KB) + comp_w (4KB) into LDS with CDNA5 async-to-LDS copies,
// then run the squeeze as a chained V_WMMA_F32_16X16X4_F32 GEMM:
//   comps1[b,c] = sum_i pooled[b,i,c] * comp_w[c,i]
// M=16 (batch), N=cols 0..3 (component), 256 K-steps of 4 with a
// block-diagonal B fragment built by v_cndmask (no divergent loads).
// Then Hamilton combine -> u1 (16x4), then elementwise excite -> u2 (16,256,4).
// ---------------------------------------------------------------------------
__global__ __launch_bounds__(32) void qca_excite(const float* __restrict__ pooled,
                                                 const float* __restrict__ comp_w,
                                                 const float* __restrict__ comp_b,
                                                 const float* __restrict__ exc_w,
                                                 const float* __restrict__ exc_b,
                                                 float* __restrict__ u2) {
  __shared__ float lds_pool[B_ * C_ * 4];   // 64KB staged pooled
  __shared__ float lds_w[4 * C_];           // 4KB staged comp_w
  __shared__ float comps[B_][4];
  __shared__ float u1s[B_][4];

  const int lane = threadIdx.x;   // 0..31
  const int half = lane >> 4;     // 0: lanes 0-15, 1: lanes 16-31
  const int m    = lane & 15;     // A row (batch) / B-D column index
  const int c0   = 2 * half;      // K index held in fragment .x
  const int c1   = c0 + 1;        // K index held in fragment .y

  // --- async-stage operands into LDS (ASYNCcnt-tracked, 16B per lane) ------
  {
    const unsigned lane16 = (unsigned)lane * 16u;
    // pooled: 64KB = 128 wave-issues of 512B
    for (unsigned it = 0; it < 128; ++it) {
      const unsigned off = it * 512u + lane16;
      const unsigned lds_addr = (unsigned)(size_t)((const char*)lds_pool + off);
      const char* g = (const char*)pooled + off;
      asm volatile("global_load_async_to_lds_b128 %0, %1, off"
                   :: "v"(lds_addr), "v"(g) : "memory");
    }
    // comp_w: 4KB = 8 wave-issues of 512B
    for (unsigned it = 0; it < 8; ++it) {
      const unsigned off = it * 512u + lane16;
      const unsigned lds_addr = (unsigned)(size_t)((const char*)lds_w + off);
      const char* g = (const char*)comp_w + off;
      asm volatile("global_load_async_to_lds_b128 %0, %1, off"
                   :: "v"(lds_addr), "v"(g) : "memory");
    }
    asm volatile("s_wait_asynccnt 0x0" ::: "memory");
  }
  __syncthreads();

  v8f acc[4];
#pragma unroll
  for (int u = 0; u < 4; ++u) acc[u] = (v8f){0.f, 0.f, 0.f, 0.f, 0.f, 0.f, 0.f, 0.f};

  // 256 K-steps of 4, interleaved over 4 accumulators for ILP.
  for (int i = 0; i < C_; i += 4) {
#pragma unroll
    for (int u = 0; u < 4; ++u) {
      const int ii = i + u;
      // A[b=m, k]: pooled[b, ii, k]; k = c0 (a.x), c1 (a.y) -> one ds_load_b64
      const v2f pv = *(const v2f*)&lds_pool[((size_t)m * C_ + ii) * 4 + c0];
      v2f a;
      a.x = pv.x;
      a.y = pv.y;
      // Unconditional LDS loads, then value-select -> v_cndmask (no branches)
      const float wv0 = lds_w[c0 * C_ + ii];
      const float wv1 = lds_w[c1 * C_ + ii];
      v2f bm;
      bm.x = (m == c0) ? wv0 : 0.0f;
      bm.y = (m == c1) ? wv1 : 0.0f;
      acc[u] = __builtin_amdgcn_wmma_f32_16x16x4_f32(
          /*neg_a=*/false, a, /*neg_b=*/false, bm,
          /*c_mod=*/(short)0, acc[u], /*reuse_a=*/false, /*reuse_b=*/false);
    }
  }
  v8f d = acc[0] + acc[1] + acc[2] + acc[3];

  // D layout: VGPR j holds M = 8*half + j at column n = m (lanes 0-15 / 16-31).
  if (m < 4) {
#pragma unroll
    for (int j = 0; j < 8; ++j) comps[half * 8 + j][m] = d[j];
  }
  __syncthreads();

  if (lane < B_) {
    const float r  = comps[lane][0] + comp_b[0];
    const float xc = comps[lane][1] + comp_b[1];
    const float yc = comps[lane][2] + comp_b[2];
    const float zc = comps[lane][3] + comp_b[3];
    u1s[lane][0] = r + xc + yc + zc;
    u1s[lane][1] = xc - r - zc + yc;
    u1s[lane][2] = yc + zc - r - xc;
    u1s[lane][3] = zc - yc + xc - r;
  }
  __syncthreads();

  // Excite (Cin = 1, purely elementwise) + second Hamilton combine.
  for (int p = lane; p < NP; p += 32) {
    const int b  = p >> 8;        // / C_
    const int ch = p & (C_ - 1);
    float c2[4];
#pragma unroll
    for (int c = 0; c < 4; ++c)
      c2[c] = u1s[b][c] * exc_w[c * C_ + ch] + exc_b[c * C_ + ch];
    const float r = c2[0], xc = c2[1], yc = c2[2], zc = c2[3];
    v4f o;
    o.x = r + xc + yc + zc;
    o.y = xc - r - zc + yc;
    o.z = yc + zc - r - xc;
    o.w = zc - yc + xc - r;
    ((v4f*)u2)[p] = o;
  }
}

// ---------------------------------------------------------------------------
// Kernel 3: out = x * u2 broadcast over spatial. Blocks walk (b,ch) pairs in
// REVERSE order so the pass starts on the ~192MB tail of x still resident in
// L2 from kernel 1. Output uses non-temporal stores so the 256MB of write
// traffic does not evict those x lines.
// ---------------------------------------------------------------------------
__global__ __launch_bounds__(256) void qca_scale(const float* __restrict__ x,
                                                 const float* __restrict__ u2,
                                                 float* __restrict__ out) {
  const int p = (NP - 1) - blockIdx.x;            // reversed traversal
  const v4f s = ((const v4f*)u2)[p];              // uniform per block
  const v4f* xp = (const v4f*)x + (size_t)p * SP;
  v4f* op = (v4f*)out + (size_t)p * SP;
#pragma unroll
  for (int i = 0; i < 16; ++i) {
    const int idx = threadIdx.x + i * 256;
    v4f v = xp[idx];
    v.x *= s.x;
    v.y *= s.y;
    v.z *= s.z;
    v.w *= s.w;
    __builtin_nontemporal_store(v, op + idx);
  }
}

// ---------------------------------------------------------------------------
extern "C" void kernel_launch(void* const* d_in, const int* in_sizes, int n_in,
                              void* d_out, int out_size, void* d_ws, size_t ws_size,
                              hipStream_t stream) {
  const float* x      = (const float*)d_in[0];  // (16,256,64,64,4)
  const float* comp_w = (const float*)d_in[1];  // (4,1,256)
  const float* comp_b = (const float*)d_in[2];  // (4,1)
  const float* exc_w  = (const float*)d_in[3];  // (4,256,1)
  const float* exc_b  = (const float*)d_in[4];  // (4,256)
  float* out = (float*)d_out;

  float* pooled = (float*)d_ws;                       // 16*256*4 floats = 64KB
  float* u2     = (float*)((char*)d_ws + (64 << 10)); // 16*256*4 floats = 64KB

  qca_maxpool<<<NP, 256, 0, stream>>>(x, pooled);
  qca_excite<<<1, 32, 0, stream>>>(pooled, comp_w, comp_b, exc_w, exc_b, u2);
  qca_scale<<<NP, 256, 0, stream>>>(x, u2, out);
}